// EnsembleModel_65403761984193
// MI455X (gfx1250) — compile-verified
//
#include <hip/hip_runtime.h>
#include <hip/hip_bf16.h>
#include <math.h>
#include <stdint.h>

// Problem constants (match the reference).
#define NPOS   1024
#define NBINS  (NPOS * NPOS)      // 2^20 bins, 4 MB fp32 — L2-resident
#define BATCH  32
#define SEQ    1024
#define ALPHA  0.3f
#define ROWS   8                  // rows of the [S,S] plane per block
#define TPB    256                // 8 wave32 per block

// Native clang vector types (required for __builtin_nontemporal_*).
typedef float nfloat4 __attribute__((ext_vector_type(4)));
typedef int   nint4   __attribute__((ext_vector_type(4)));

// ---------------------------------------------------------------------------
// CDNA5 helpers
// ---------------------------------------------------------------------------

// Native no-return fp32 atomic add (GVS mode: SGPR base + signed VGPR offset).
// Fire-and-forget: tracked with STOREcnt, resolved in L2's atomic units.
__device__ __forceinline__ void atomic_add_f32_noret(float* base, int bin, float val) {
    asm volatile("global_atomic_add_f32 %0, %1, %2 scope:SCOPE_DEV"
                 :
                 : "v"(bin * 4), "v"(val), "s"(base));
}

// Async DMA of one 16B chunk per lane from global into LDS (ASYNCcnt path),
// then drain ASYNCcnt and barrier so all waves see the staged row.
// LDS byte address = low 32 bits of the generic pointer (LDS aperture).
__device__ __forceinline__ void stage_row_async(const int* __restrict__ grow,
                                                int* lds, int t) {
    const unsigned lds_off = (unsigned)(uintptr_t)lds + (unsigned)t * 16u;
    const int*     gp      = grow + t * 4;
    asm volatile("global_load_async_to_lds_b128 %0, %1, off"
                 :
                 : "v"(lds_off), "v"(gp)
                 : "memory");
    asm volatile("s_wait_asynccnt 0x0" ::: "memory");
    __syncthreads();
}

// ---------------------------------------------------------------------------
// K0: zero the histogram workspace (d_ws is poisoned by the harness).
// ---------------------------------------------------------------------------
__global__ void k_zero_hist(float* __restrict__ hist) {
    const int i = (blockIdx.x * TPB + threadIdx.x) * 4;
    nfloat4 z = {0.0f, 0.0f, 0.0f, 0.0f};
    *(nfloat4*)(hist + i) = z;
}

// ---------------------------------------------------------------------------
// K1: scatter-add  hist[adds[b,p1]*NPOS + adds[b,p2]] += a_arc[b,p1,p2]
// One block = one (batch, 8-row tile). adds row staged in LDS via async DMA.
// a_arc streamed with non-temporal b128 loads; updates are native no-return
// global_atomic_add_f32 that resolve in L2 (bins span only 4 MB).
// ---------------------------------------------------------------------------
__global__ void k_scatter(const float* __restrict__ a_arc,
                          const int*   __restrict__ adds,
                          float*       __restrict__ hist) {
    __shared__ __attribute__((aligned(16))) int s_idx[SEQ];
    const int b    = blockIdx.x / (SEQ / ROWS);
    const int tile = blockIdx.x % (SEQ / ROWS);
    const int t    = threadIdx.x;

    stage_row_async(adds + b * SEQ, s_idx, t);

    const int row0 = tile * ROWS;
    const int c4   = t * 4;
    for (int r = 0; r < ROWS; ++r) {
        const int    row  = row0 + r;
        const int    base = s_idx[row] * NPOS;          // adds[b,p1] * NPOS
        const float* ap   = a_arc + ((size_t)(b * SEQ + row)) * SEQ;

        // gfx1250 prefetch path (global_prefetch_b8) for the next row stream.
        __builtin_prefetch(ap + SEQ, 0, 0);

        // Non-temporal 16B load: don't pollute L2 (histogram lives there).
        nfloat4 av = __builtin_nontemporal_load(((const nfloat4*)ap) + t);

#pragma unroll
        for (int j = 0; j < 4; ++j) {
            const int bin = base + s_idx[c4 + j];       // always < NBINS
            atomic_add_f32_noret(hist, bin, av[j]);
        }
    }
}

// ---------------------------------------------------------------------------
// K2: score = sigmoid(hist), in place over the 1M-bin table.
// ---------------------------------------------------------------------------
__global__ void k_sigmoid(float* __restrict__ hist) {
    const int i = (blockIdx.x * TPB + threadIdx.x) * 4;
    nfloat4 h = *(const nfloat4*)(hist + i);
    nfloat4 s;
#pragma unroll
    for (int j = 0; j < 4; ++j) {
        s[j] = 1.0f / (1.0f + expf(-h[j]));
    }
    *(nfloat4*)(hist + i) = s;
}

// ---------------------------------------------------------------------------
// K3: out[b,i,j] = s_arc[b,i,j] + ALPHA * score[pos[b,i]*NPOS + pos[b,j]]
// pos row staged in LDS via async DMA; s_arc / out streamed non-temporally;
// score gathers stay regular-temporal so the 4 MB table remains L2-resident.
// ---------------------------------------------------------------------------
__global__ void k_gather(const float* __restrict__ s_arc,
                         const int*   __restrict__ pos,
                         const float* __restrict__ score,
                         float*       __restrict__ out) {
    __shared__ __attribute__((aligned(16))) int s_pos[SEQ];
    const int b    = blockIdx.x / (SEQ / ROWS);
    const int tile = blockIdx.x % (SEQ / ROWS);
    const int t    = threadIdx.x;

    stage_row_async(pos + b * SEQ, s_pos, t);

    const int row0 = tile * ROWS;
    const int c4   = t * 4;
    for (int r = 0; r < ROWS; ++r) {
        const int    row  = row0 + r;
        const size_t off  = ((size_t)(b * SEQ + row)) * SEQ;
        const int    base = s_pos[row] * NPOS;

        __builtin_prefetch(s_arc + off + SEQ, 0, 0);

        nfloat4 sv = __builtin_nontemporal_load(((const nfloat4*)(s_arc + off)) + t);
        nfloat4 o;
#pragma unroll
        for (int j = 0; j < 4; ++j) {
            const int idx = base + s_pos[c4 + j];
            // Faithful to the reference's validity mask (always true for
            // in-spec inputs, nearly free to keep).
            const float g = ((unsigned)idx < (unsigned)NBINS) ? score[idx] : 0.0f;
            o[j] = sv[j] + ALPHA * g;
        }
        __builtin_nontemporal_store(o, ((nfloat4*)(out + off)) + t);
    }
}

// ---------------------------------------------------------------------------
// Launch: zero -> scatter -> sigmoid -> gather (ordering via stream).
// ---------------------------------------------------------------------------
extern "C" void kernel_launch(void* const* d_in, const int* in_sizes, int n_in,
                              void* d_out, int out_size, void* d_ws, size_t ws_size,
                              hipStream_t stream) {
    (void)in_sizes; (void)n_in; (void)out_size; (void)ws_size;

    const float* s_arc = (const float*)d_in[0];  // [B,S,S] fp32
    const float* a_arc = (const float*)d_in[1];  // [B,S,S] fp32
    const int*   adds  = (const int*)d_in[2];    // [B,S] i32
    const int*   pos   = (const int*)d_in[3];    // [B,S] i32
    float*       out   = (float*)d_out;          // [B,S,S] fp32
    float*       hist  = (float*)d_ws;           // 4 MB scratch (NBINS fp32)

    const int tile_blocks = BATCH * (SEQ / ROWS);   // 4096
    const int bin_blocks  = NBINS / (TPB * 4);      // 1024

    k_zero_hist<<<bin_blocks,  TPB, 0, stream>>>(hist);
    k_scatter  <<<tile_blocks, TPB, 0, stream>>>(a_arc, adds, hist);
    k_sigmoid  <<<bin_blocks,  TPB, 0, stream>>>(hist);
    k_gather   <<<tile_blocks, TPB, 0, stream>>>(s_arc, pos, hist, out);
}